// SMNetEncoder_80307298501389
// MI455X (gfx1250) — compile-verified
//
#include <hip/hip_runtime.h>

#define D 256
#define ED 32
#define TM 64
#define LDS_STRIDE (D + 4)   // 260 ; 260 % 64 == 4 -> conflict-free dual-half b64 A-frag reads

typedef __attribute__((ext_vector_type(2))) float v2f;
typedef __attribute__((ext_vector_type(4))) float v4f;
typedef __attribute__((ext_vector_type(8))) float v8f;

// f32 WMMA: D(16x16 f32) = A(16x4 f32) * B(4x16 f32) + C
static __device__ __forceinline__ v8f wmma_f32x4(v2f a, v2f b, v8f c) {
  return __builtin_amdgcn_wmma_f32_16x16x4_f32(false, a, false, b, (short)0, c, false, false);
}

static __device__ __forceinline__ v4f relu4(v4f v) {
  v.x = v.x > 0.f ? v.x : 0.f;
  v.y = v.y > 0.f ? v.y : 0.f;
  v.z = v.z > 0.f ? v.z : 0.f;
  v.w = v.w > 0.f ? v.w : 0.f;
  return v;
}

// ---------------------------------------------------------------------------
// agg = (1+eps) * y   (replaces zero-fill; node kernel then reads agg only)
// ---------------------------------------------------------------------------
__global__ __launch_bounds__(256) void init_agg_kernel(
    const float* __restrict__ y, const float* __restrict__ epsv,
    float* __restrict__ agg, int n4) {
  const float ope = 1.0f + epsv[0];
  int i = blockIdx.x * blockDim.x + threadIdx.x;
  const int stride = gridDim.x * blockDim.x;
  for (; i < n4; i += stride) ((v4f*)agg)[i] = ope * ((const v4f*)y)[i];
}

// ---------------------------------------------------------------------------
// Edge pipeline: e = edge_attr @ eW + eb ; m = relu(y[src] + e) ; agg[dst] += m
// One block = 16 edges x 256 cols. Each wave: two 16x16 tiles, K=32 via WMMA.
// Gather loads issued BEFORE the WMMA chain so L2 latency hides under matmul.
// ---------------------------------------------------------------------------
__global__ __launch_bounds__(256) void edge_kernel(
    const float* __restrict__ y, const int* __restrict__ src, const int* __restrict__ dst,
    const float* __restrict__ eattr, const float* __restrict__ eW, const float* __restrict__ eb,
    float* __restrict__ agg, int E_)
{
  __shared__ float sA[16][ED + 2];   // stride 34: conflict-free A-frag reads
  __shared__ int   sIdx[32];         // [0,16): src ; [16,32): dst
  const int t  = threadIdx.x;
  const int e0 = blockIdx.x * 16;
  const bool full = (e0 + 16 <= E_);

  if (t < 32) {
    const int m = t & 15;
    const int e = e0 + m;
    sIdx[t] = (e < E_) ? ((t < 16) ? src[e] : dst[e]) : 0;
  }
  {
    const int idx = t * 2;
    const int r = idx >> 5, c = idx & 31;
    const int e = e0 + r;
    v2f av = {0.f, 0.f};
    if (e < E_) av = *(const v2f*)(eattr + (long)e * ED + c);
    *(v2f*)&sA[r][c] = av;
  }
  __syncthreads();

  const int wave = t >> 5, lane = t & 31;
  const int laneHi = lane >> 4, lane16 = lane & 15;

  // per-lane row byte offsets for the 8 C rows this lane owns (reused across col tiles)
  unsigned offS[8], offD[8];
#pragma unroll
  for (int i = 0; i < 8; ++i) {
    const int m = i + 8 * laneHi;
    offS[i] = ((unsigned)sIdx[m]) * (unsigned)(D * 4);
    offD[i] = ((unsigned)sIdx[16 + m]) * (unsigned)(D * 4);
  }

  for (int nt = wave; nt < D / 16; nt += 8) {
    const int col = nt * 16 + lane16;
    const unsigned colB = (unsigned)col << 2;

    // 1) issue all 8 independent gathers up front (distinct registers)
    float gth[8];
    if (full) {
#pragma unroll
      for (int i = 0; i < 8; ++i)
        gth[i] = *(const float*)((const char*)y + (offS[i] + colB));
    } else {
#pragma unroll
      for (int i = 0; i < 8; ++i) {
        const int e = e0 + i + 8 * laneHi;
        gth[i] = (e < E_) ? *(const float*)((const char*)y + (offS[i] + colB)) : 0.f;
      }
    }

    // 2) WMMA chain (gather latency hides under these)
    const float bias = eb[col];
    v8f acc;
#pragma unroll
    for (int i = 0; i < 8; ++i) acc[i] = bias;       // fold eb into C
#pragma unroll
    for (int k = 0; k < ED; k += 4) {
      v2f a, b;
      a.x = sA[lane16][k + 2 * laneHi];              // A: lanes0-15 K=k..k+1, lanes16-31 K=k+2..k+3
      a.y = sA[lane16][k + 2 * laneHi + 1];
      b.x = eW[(long)(k + laneHi) * D + col];        // B: row-major eW[K][col]
      b.y = eW[(long)(k + 2 + laneHi) * D + col];
      acc = wmma_f32x4(a, b, acc);
    }

    // 3) combine + back-to-back atomics (no per-element waits)
    // C layout: vgpr i, lane16 = col, row = i + 8*laneHi
    if (full) {
#pragma unroll
      for (int i = 0; i < 8; ++i) {
        float v = acc[i] + gth[i];
        v = v > 0.f ? v : 0.f;
        atomicAdd((float*)((char*)agg + (offD[i] + colB)), v);
      }
    } else {
#pragma unroll
      for (int i = 0; i < 8; ++i) {
        const int e = e0 + i + 8 * laneHi;
        if (e < E_) {
          float v = acc[i] + gth[i];
          v = v > 0.f ? v : 0.f;
          atomicAdd((float*)((char*)agg + (offD[i] + colB)), v);
        }
      }
    }
  }
}

// ---------------------------------------------------------------------------
// Node MLP: z = agg (= (1+eps)*y + scatter) ; u = relu(z@W1+b1) ;
//           hout = base + u@W2 + b2
// One block = TM(=64) rows x 256 cols. Two LDS-staged WMMA GEMMs,
// vectorized b128 load/epilogue.
// ---------------------------------------------------------------------------
__global__ __launch_bounds__(256) void node_mlp_kernel(
    const float* __restrict__ zin, const float* __restrict__ base,
    const float* __restrict__ W1, const float* __restrict__ b1,
    const float* __restrict__ W2, const float* __restrict__ b2,
    float* __restrict__ hout, int N_)
{
  __shared__ float sZ[TM][LDS_STRIDE];
  __shared__ float sU[TM][LDS_STRIDE];
  const int t  = threadIdx.x;
  const int r0 = blockIdx.x * TM;

  for (int i = t; i < TM * D / 4; i += 256) {
    const int idx = i * 4, r = idx / D, c = idx % D;
    const int g = r0 + r;
    v4f z = {0.f, 0.f, 0.f, 0.f};
    if (g < N_) z = *(const v4f*)(zin + (long)g * D + c);
    *(v4f*)&sZ[r][c] = z;
  }
  __syncthreads();

  const int wave = t >> 5, lane = t & 31;
  const int laneHi = lane >> 4, lane16 = lane & 15;

  // GEMM1: sU = relu(sZ @ W1 + b1)
  for (int tt = wave; tt < (TM / 16) * (D / 16); tt += 8) {
    const int rt = tt >> 4, ct = tt & 15;
    const int arow = rt * 16 + lane16;
    const int col  = ct * 16 + lane16;
    const float bias = b1[col];
    v8f acc;
#pragma unroll
    for (int i = 0; i < 8; ++i) acc[i] = bias;
#pragma unroll 8
    for (int k = 0; k < D; k += 4) {
      v2f a, b;
      a.x = sZ[arow][k + 2 * laneHi];
      a.y = sZ[arow][k + 2 * laneHi + 1];
      b.x = W1[(long)(k + laneHi) * D + col];
      b.y = W1[(long)(k + 2 + laneHi) * D + col];
      acc = wmma_f32x4(a, b, acc);
    }
#pragma unroll
    for (int i = 0; i < 8; ++i) {
      const int row = rt * 16 + i + 8 * laneHi;
      const float v = acc[i];
      sU[row][col] = v > 0.f ? v : 0.f;
    }
  }
  __syncthreads();

  // GEMM2: sZ = sU @ W2 + b2   (sZ dead after GEMM1 -> reuse as output stage)
  for (int tt = wave; tt < (TM / 16) * (D / 16); tt += 8) {
    const int rt = tt >> 4, ct = tt & 15;
    const int arow = rt * 16 + lane16;
    const int col  = ct * 16 + lane16;
    const float bias = b2[col];
    v8f acc;
#pragma unroll
    for (int i = 0; i < 8; ++i) acc[i] = bias;
#pragma unroll 8
    for (int k = 0; k < D; k += 4) {
      v2f a, b;
      a.x = sU[arow][k + 2 * laneHi];
      a.y = sU[arow][k + 2 * laneHi + 1];
      b.x = W2[(long)(k + laneHi) * D + col];
      b.y = W2[(long)(k + 2 + laneHi) * D + col];
      acc = wmma_f32x4(a, b, acc);
    }
#pragma unroll
    for (int i = 0; i < 8; ++i) {
      const int row = rt * 16 + i + 8 * laneHi;
      sZ[row][col] = acc[i];
    }
  }
  __syncthreads();

  // vectorized epilogue: hout = sZ (+ base), b128 traffic
  for (int i = t; i < TM * D / 4; i += 256) {
    const int idx = i * 4, r = idx / D, c = idx % D;
    const int g = r0 + r;
    if (g < N_) {
      v4f v = *(const v4f*)&sZ[r][c];
      if (base) v += *(const v4f*)(base + (long)g * D + c);
      *(v4f*)(hout + (long)g * D + c) = v;
    }
  }
}

// ---------------------------------------------------------------------------
// y = relu(layernorm(h) * gamma + beta) ; one wave per row, b128 vector I/O
// ---------------------------------------------------------------------------
__global__ __launch_bounds__(256) void ln_relu_kernel(
    const float* __restrict__ h, const float* __restrict__ g, const float* __restrict__ b,
    float* __restrict__ out, int N_)
{
  const int wave = threadIdx.x >> 5, lane = threadIdx.x & 31;
  const int row = blockIdx.x * 8 + wave;
  if (row >= N_) return;
  const float* x = h + (long)row * D;
  const int c0 = lane * 4;
  const v4f u0 = *(const v4f*)(x + c0);
  const v4f u1 = *(const v4f*)(x + 128 + c0);
  float s = (u0.x + u0.y) + (u0.z + u0.w) + (u1.x + u1.y) + (u1.z + u1.w);
#pragma unroll
  for (int off = 16; off > 0; off >>= 1) s += __shfl_xor(s, off, 32);
  const float mu = s * (1.0f / D);
  v4f d0 = u0 - mu, d1 = u1 - mu;
  float q = (d0.x * d0.x + d0.y * d0.y) + (d0.z * d0.z + d0.w * d0.w) +
            (d1.x * d1.x + d1.y * d1.y) + (d1.z * d1.z + d1.w * d1.w);
#pragma unroll
  for (int off = 16; off > 0; off >>= 1) q += __shfl_xor(q, off, 32);
  const float inv = rsqrtf(q * (1.0f / D) + 1e-5f);
  const v4f g0 = *(const v4f*)(g + c0);
  const v4f g1 = *(const v4f*)(g + 128 + c0);
  const v4f b0 = *(const v4f*)(b + c0);
  const v4f b1v = *(const v4f*)(b + 128 + c0);
  float* o = out + (long)row * D;
  *(v4f*)(o + c0)       = relu4(d0 * inv * g0 + b0);
  *(v4f*)(o + 128 + c0) = relu4(d1 * inv * g1 + b1v);
}

// ---------------------------------------------------------------------------
extern "C" void kernel_launch(void* const* d_in, const int* in_sizes, int n_in,
                              void* d_out, int out_size, void* d_ws, size_t ws_size,
                              hipStream_t stream) {
  const float* x     = (const float*)d_in[0];
  const int*   eidx  = (const int*)d_in[1];
  const float* eattr = (const float*)d_in[2];
  const float* W1    = (const float*)d_in[3];
  const float* b1    = (const float*)d_in[4];
  const float* W2    = (const float*)d_in[5];
  const float* b2    = (const float*)d_in[6];
  const float* eW    = (const float*)d_in[7];
  const float* eb    = (const float*)d_in[8];
  const float* eps   = (const float*)d_in[9];
  const float* gamma = (const float*)d_in[10];
  const float* beta  = (const float*)d_in[11];

  const int N = in_sizes[0] / D;
  const int E = in_sizes[1] / 2;
  const int L = in_sizes[9];
  const int* src = eidx;
  const int* dst = eidx + E;

  float* h   = (float*)d_ws;                 // [N,D]
  float* y   = h + (size_t)N * D;            // [N,D]
  float* agg = y + (size_t)N * D;            // [N,D]

  const int nd4 = N * D / 4;
  const dim3 blk(256);
  const int edgeBlocks = (E + 15) / 16;
  const int nodeBlocks = (N + TM - 1) / TM;
  const int lnBlocks   = (N + 7) / 8;

  for (int l = 0; l < L; ++l) {
    const float* yin   = x;
    const float* basep = nullptr;
    if (l > 0) {
      ln_relu_kernel<<<lnBlocks, blk, 0, stream>>>(h, gamma + (size_t)l * D,
                                                   beta + (size_t)l * D, y, N);
      yin = y;
      basep = h;
    }
    init_agg_kernel<<<1024, blk, 0, stream>>>(yin, eps + l, agg, nd4);
    edge_kernel<<<edgeBlocks, blk, 0, stream>>>(yin, src, dst, eattr,
        eW + (size_t)l * ED * D, eb + (size_t)l * D, agg, E);
    node_mlp_kernel<<<nodeBlocks, blk, 0, stream>>>(agg, basep,
        W1 + (size_t)l * D * D, b1 + (size_t)l * D,
        W2 + (size_t)l * D * D, b2 + (size_t)l * D,
        h, N);
  }
  ln_relu_kernel<<<lnBlocks, blk, 0, stream>>>(h, gamma, beta, (float*)d_out, N);
}